// HeteroRGCN_56530359550632
// MI455X (gfx1250) — compile-verified
//
#include <hip/hip_runtime.h>
#include <cstdint>
#include <cstddef>

#define HIDDEN 128
#define LN_EPS 1e-5f

typedef __attribute__((ext_vector_type(2))) float v2f;
typedef __attribute__((ext_vector_type(8))) float v8f;
typedef int v4i __attribute__((vector_size(16)));

#if defined(__has_builtin)
#if __has_builtin(__builtin_amdgcn_global_load_async_to_lds_b128)
#define HAVE_ASYNC_LDS 1
#endif
#endif
#ifndef HAVE_ASYNC_LDS
#define HAVE_ASYNC_LDS 0
#endif

// D = A(16x4 f32) * B(4x16 f32) + C(16x16 f32) on one wave32.
__device__ __forceinline__ v8f wmma_f32x4(v2f a, v2f b, v8f c) {
  return __builtin_amdgcn_wmma_f32_16x16x4_f32(
      /*neg_a=*/false, a, /*neg_b=*/false, b,
      /*c_mod=*/(short)0, c, /*reuse_a=*/false, /*reuse_b=*/false);
}

// ---------------------------------------------------------------------------
// Projection: H[N x 128] = relu(X[N x din] @ W[din x 128] + bias)
// One wave computes a 16-row x 128-col strip (8 WMMA accumulators).
// Main K loop is completely branchless; ragged K tail is handled once with
// clamped indices + A-side masking (v_cndmask, no EXEC divergence).
// ---------------------------------------------------------------------------
__global__ void __launch_bounds__(128)
proj_kernel(const float* __restrict__ X, const float* __restrict__ W,
            const float* __restrict__ bias, float* __restrict__ H,
            int N, int din) {
  const int wave = threadIdx.x >> 5;
  const int lane = threadIdx.x & 31;
  const int half = lane >> 4;
  const int lr   = lane & 15;
  const int row0 = (blockIdx.x * 4 + wave) * 16;
  if (row0 >= N) return;                       // wave-uniform exit

  const int rowA = min(row0 + lr, N - 1);
  const float* __restrict__ xr = X + (size_t)rowA * din;
  v8f acc[8] = {};

  const int kfull = din & ~3;
  for (int k0 = 0; k0 < kfull; k0 += 4) {
    const int k = k0 + 2 * half;               // lanes 0-15 -> K={0,1}, 16-31 -> K={2,3}
    v2f a;
    a[0] = xr[k];
    a[1] = xr[k + 1];
    const float* __restrict__ w0 = W + (size_t)k * HIDDEN;
#pragma unroll
    for (int n = 0; n < 8; ++n) {
      const int col = n * 16 + lr;
      v2f b;
      b[0] = w0[col];
      b[1] = w0[HIDDEN + col];
      acc[n] = wmma_f32x4(a, b, acc[n]);
    }
  }
  if (kfull < din) {                           // single ragged tail step
    const int k   = kfull + 2 * half;
    const int k0c = min(k, din - 1);
    const int k1c = min(k + 1, din - 1);
    const float m0 = (k     < din) ? 1.0f : 0.0f;
    const float m1 = (k + 1 < din) ? 1.0f : 0.0f;
    v2f a;
    a[0] = xr[k0c] * m0;                       // masking A is sufficient: 0*B = 0
    a[1] = xr[k1c] * m1;
#pragma unroll
    for (int n = 0; n < 8; ++n) {
      const int col = n * 16 + lr;
      v2f b;
      b[0] = W[(size_t)k0c * HIDDEN + col];
      b[1] = W[(size_t)k1c * HIDDEN + col];
      acc[n] = wmma_f32x4(a, b, acc[n]);
    }
  }

  if (row0 + 16 <= N) {                        // uniform full-tile store path
#pragma unroll
    for (int n = 0; n < 8; ++n) {
      const int col = n * 16 + lr;
#pragma unroll
      for (int r = 0; r < 8; ++r) {
        const int row = row0 + r + 8 * half;
        H[(size_t)row * HIDDEN + col] = fmaxf(acc[n][r] + bias[col], 0.0f);
      }
    }
  } else {
#pragma unroll
    for (int n = 0; n < 8; ++n) {
      const int col = n * 16 + lr;
#pragma unroll
      for (int r = 0; r < 8; ++r) {
        const int row = row0 + r + 8 * half;
        if (row < N)
          H[(size_t)row * HIDDEN + col] = fmaxf(acc[n][r] + bias[col], 0.0f);
      }
    }
  }
}

// ---------------------------------------------------------------------------
// Edge scatter: one wave per edge. sums[dst] += h[src]; cnt[dst] += 1.
// Gather as b128, scatter via global_atomic_add_f32.
// ---------------------------------------------------------------------------
__global__ void __launch_bounds__(256)
scatter_kernel(const int* __restrict__ src, const int* __restrict__ dst, int E,
               const float* __restrict__ hsrc, float* __restrict__ sums,
               float* __restrict__ cnt) {
  const int wid  = (int)(((size_t)blockIdx.x * blockDim.x + threadIdx.x) >> 5);
  const int lane = threadIdx.x & 31;
  if (wid >= E) return;
  const int s = src[wid];
  const int d = dst[wid];
  const float4 v = ((const float4*)(hsrc + (size_t)s * HIDDEN))[lane];
  float* sm = sums + (size_t)d * HIDDEN + lane * 4;
  atomicAdd(&sm[0], v.x);
  atomicAdd(&sm[1], v.y);
  atomicAdd(&sm[2], v.z);
  atomicAdd(&sm[3], v.w);
  if (lane == 0) atomicAdd(&cnt[d], 1.0f);
}

// ---------------------------------------------------------------------------
// SAGEConv dense part, fused:
//   accum[N x 128] += (sums/cnt) @ Wl + bl + hdst @ Wr
// Wl/Wr (64 KB each) are staged into LDS once per workgroup via the gfx1250
// async global->LDS path (fallback: plain copy). B-fragments then come from
// ds_load; A-fragments are b64 global loads.
// ---------------------------------------------------------------------------
__global__ void __launch_bounds__(128)
sage_kernel(const float* __restrict__ sums, const float* __restrict__ cnt,
            const float* __restrict__ hdst,
            const float* __restrict__ Wl, const float* __restrict__ blv,
            const float* __restrict__ Wr,
            float* __restrict__ accum, int N) {
  extern __shared__ float smem[];              // [2 * 128 * 128] = 128 KB
  const int tid  = threadIdx.x;
  const int wave = tid >> 5;
  const int lane = tid & 31;
  const int half = lane >> 4;
  const int lr   = lane & 15;
  const int row0 = (blockIdx.x * 4 + wave) * 16;

  // ---- stage Wl || Wr into LDS (all 128 threads participate) ----
#if HAVE_ASYNC_LDS
#pragma unroll 4
  for (int i = tid; i < 4096; i += 128) {      // 4096 float4 per matrix
    __builtin_amdgcn_global_load_async_to_lds_b128(
        (__attribute__((address_space(1))) v4i*)(Wl + 4 * i),
        (__attribute__((address_space(3))) v4i*)(smem + 4 * i), 0, 0);
  }
#pragma unroll 4
  for (int i = tid; i < 4096; i += 128) {
    __builtin_amdgcn_global_load_async_to_lds_b128(
        (__attribute__((address_space(1))) v4i*)(Wr + 4 * i),
        (__attribute__((address_space(3))) v4i*)(smem + 16384 + 4 * i), 0, 0);
  }
  asm volatile("s_wait_asynccnt 0x0" ::: "memory");
  __syncthreads();
#else
  for (int i = tid; i < 4096; i += 128) {
    ((float4*)smem)[i]           = ((const float4*)Wl)[i];
    ((float4*)(smem + 16384))[i] = ((const float4*)Wr)[i];
  }
  __syncthreads();
#endif

  if (row0 >= N) return;                       // wave-uniform (after barrier)

  const int rowA   = min(row0 + lr, N - 1);
  const float invc = 1.0f / fmaxf(cnt[rowA], 1.0f);
  const float* __restrict__ sr = sums + (size_t)rowA * HIDDEN;
  const float* __restrict__ hr = hdst + (size_t)rowA * HIDDEN;

  v8f acc[8] = {};
  for (int k0 = 0; k0 < HIDDEN; k0 += 4) {
    const int k = k0 + 2 * half;
    v2f am = *(const v2f*)(sr + k);            // 8B aligned (k even)
    v2f ah = *(const v2f*)(hr + k);
    am[0] *= invc; am[1] *= invc;
    const float* l0 = smem + k * HIDDEN;
    const float* r0 = smem + 16384 + k * HIDDEN;
#pragma unroll
    for (int n = 0; n < 8; ++n) {
      const int col = n * 16 + lr;
      v2f bl_, br_;
      bl_[0] = l0[col];
      bl_[1] = l0[HIDDEN + col];
      br_[0] = r0[col];
      br_[1] = r0[HIDDEN + col];
      acc[n] = wmma_f32x4(am, bl_, acc[n]);
      acc[n] = wmma_f32x4(ah, br_, acc[n]);
    }
  }

  if (row0 + 16 <= N) {
#pragma unroll
    for (int n = 0; n < 8; ++n) {
      const int col = n * 16 + lr;
#pragma unroll
      for (int r = 0; r < 8; ++r) {
        const int row = row0 + r + 8 * half;
        accum[(size_t)row * HIDDEN + col] += acc[n][r] + blv[col];
      }
    }
  } else {
#pragma unroll
    for (int n = 0; n < 8; ++n) {
      const int col = n * 16 + lr;
#pragma unroll
      for (int r = 0; r < 8; ++r) {
        const int row = row0 + r + 8 * half;
        if (row < N)
          accum[(size_t)row * HIDDEN + col] += acc[n][r] + blv[col];
      }
    }
  }
}

// ---------------------------------------------------------------------------
// LayerNorm + ReLU, one wave per row (wave32 shuffle reduction).
// ---------------------------------------------------------------------------
__global__ void __launch_bounds__(256)
ln_relu_kernel(const float* __restrict__ x, const float* __restrict__ g,
               const float* __restrict__ b, float* __restrict__ out, int N) {
  const int wid  = (int)(((size_t)blockIdx.x * blockDim.x + threadIdx.x) >> 5);
  const int lane = threadIdx.x & 31;
  if (wid >= N) return;
  const float4 vv = ((const float4*)(x + (size_t)wid * HIDDEN))[lane];
  float v[4] = {vv.x, vv.y, vv.z, vv.w};
  float s = v[0] + v[1] + v[2] + v[3];
#pragma unroll
  for (int off = 16; off; off >>= 1) s += __shfl_xor(s, off, 32);
  const float mu = s * (1.0f / HIDDEN);
  float var = 0.0f;
#pragma unroll
  for (int i = 0; i < 4; ++i) { const float d = v[i] - mu; var += d * d; }
#pragma unroll
  for (int off = 16; off; off >>= 1) var += __shfl_xor(var, off, 32);
  const float rs = rsqrtf(var * (1.0f / HIDDEN) + LN_EPS);
#pragma unroll
  for (int i = 0; i < 4; ++i) {
    const int c = lane * 4 + i;
    const float o = (v[i] - mu) * rs * g[c] + b[c];
    out[(size_t)wid * HIDDEN + c] = fmaxf(o, 0.0f);
  }
}

// ---------------------------------------------------------------------------
// Classifier: out[i] = dot(h[i], W) + b   (one wave per row)
// ---------------------------------------------------------------------------
__global__ void __launch_bounds__(256)
clf_kernel(const float* __restrict__ h, const float* __restrict__ W,
           const float* __restrict__ b, float* __restrict__ out, int N) {
  const int wid  = (int)(((size_t)blockIdx.x * blockDim.x + threadIdx.x) >> 5);
  const int lane = threadIdx.x & 31;
  if (wid >= N) return;
  const float4 xv = ((const float4*)(h + (size_t)wid * HIDDEN))[lane];
  const float4 wv = ((const float4*)W)[lane];
  float s = xv.x * wv.x + xv.y * wv.y + xv.z * wv.z + xv.w * wv.w;
#pragma unroll
  for (int off = 16; off; off >>= 1) s += __shfl_xor(s, off, 32);
  if (lane == 0) out[wid] = s + b[0];
}

__global__ void zero_kernel(float* __restrict__ p, size_t n) {
  size_t i = (size_t)blockIdx.x * blockDim.x + threadIdx.x;
  const size_t stride = (size_t)gridDim.x * blockDim.x;
  for (; i < n; i += stride) p[i] = 0.0f;
}

// ---------------------------------------------------------------------------
// Host orchestration
// ---------------------------------------------------------------------------
extern "C" void kernel_launch(void* const* d_in, const int* in_sizes, int n_in,
                              void* d_out, int out_size, void* d_ws, size_t ws_size,
                              hipStream_t stream) {
  (void)n_in; (void)out_size; (void)ws_size;
  constexpr int NT = 7;
  static const int counts[NT] = {60000, 100000, 300000, 10000, 40000, 20000, 1000};
  static const int dins[NT]   = {774, 778, 7, 6, 2, 3, 4};
  // edge (src type, dst type) per EDGE_SPECS order
  static const int esrc[14] = {0, 2, 2, 1, 0, 1, 0, 3, 0, 4, 0, 5, 0, 6};
  static const int edst[14] = {2, 0, 1, 2, 1, 0, 3, 0, 4, 0, 5, 0, 6, 0};

  size_t off[NT + 1];
  off[0] = 0;
  for (int t = 0; t < NT; ++t) off[t + 1] = off[t] + (size_t)counts[t];
  const size_t NTOT = off[NT];  // 531000 rows

  // Workspace layout (floats): h | accum | sums | cnt  (~818 MB total)
  float* ws    = (float*)d_ws;
  float* hbuf  = ws;
  float* accum = hbuf  + NTOT * HIDDEN;
  float* sums  = accum + NTOT * HIDDEN;
  float* cnt   = sums  + NTOT * HIDDEN;

  // Input index map (setup_inputs dict order):
  // 0..6   xs, 7..20 edges, 21..34 proj {W,b}x7, 35..76 conv1 {Wl,bl,Wr}x14,
  // 77..118 conv2, 119..132 ln1 {g,b}x7, 133..146 ln2, 147 clfW, 148 clfb.

  // ---- Projection ----
  for (int t = 0; t < NT; ++t) {
    const int N = counts[t];
    proj_kernel<<<dim3((N + 63) / 64), 128, 0, stream>>>(
        (const float*)d_in[t], (const float*)d_in[21 + 2 * t],
        (const float*)d_in[22 + 2 * t], hbuf + off[t] * HIDDEN, N, dins[t]);
  }

  // ---- Two hetero layers ----
  for (int layer = 0; layer < 2; ++layer) {
    const int convbase = layer ? 77 : 35;
    const int lnbase   = layer ? 133 : 119;

    // zero accum, sums, cnt (contiguous region)
    const size_t zn = 2 * NTOT * HIDDEN + NTOT;
    zero_kernel<<<2048, 256, 0, stream>>>(accum, zn);

    // gather + segment-sum per edge type
    for (int e = 0; e < 14; ++e) {
      const int* ei = (const int*)d_in[7 + e];
      const int  E  = in_sizes[7 + e] / 2;   // [2,E] -> src then dst
      scatter_kernel<<<dim3((E + 7) / 8), 256, 0, stream>>>(
          ei, ei + E, E,
          hbuf + off[esrc[e]] * HIDDEN,
          sums + off[edst[e]] * HIDDEN,
          cnt  + off[edst[e]]);
    }

    // dense SAGE part per edge type (serialized RMW accumulation into accum)
    for (int e = 0; e < 14; ++e) {
      const int d = edst[e];
      const int N = counts[d];
      sage_kernel<<<dim3((N + 63) / 64), 128, 2 * HIDDEN * HIDDEN * sizeof(float),
                    stream>>>(
          sums + off[d] * HIDDEN, cnt + off[d], hbuf + off[d] * HIDDEN,
          (const float*)d_in[convbase + 3 * e],
          (const float*)d_in[convbase + 3 * e + 1],
          (const float*)d_in[convbase + 3 * e + 2],
          accum + off[d] * HIDDEN, N);
    }

    // LayerNorm + ReLU -> next h
    for (int t = 0; t < NT; ++t) {
      const int N = counts[t];
      ln_relu_kernel<<<dim3(((size_t)N * 32 + 255) / 256), 256, 0, stream>>>(
          accum + off[t] * HIDDEN,
          (const float*)d_in[lnbase + 2 * t],
          (const float*)d_in[lnbase + 2 * t + 1],
          hbuf + off[t] * HIDDEN, N);
    }
  }

  // ---- Classifier over commit rows ----
  clf_kernel<<<dim3(((size_t)counts[0] * 32 + 255) / 256), 256, 0, stream>>>(
      hbuf, (const float*)d_in[147], (const float*)d_in[148],
      (float*)d_out, counts[0]);
}